// ESNBlock_45054206935149
// MI455X (gfx1250) — compile-verified
//
#include <hip/hip_runtime.h>
#include <math.h>
#include <stdint.h>

// ESN forward for MI455X (gfx1250).
// Phase 1: u_in = x @ W_in^T + b  via V_WMMA_F32_16X16X4_F32, written to d_out.
// Phase 2: persistent 32-WG kernel; each WG pins a 32-col slice of W (128KB)
//          in LDS (CDNA5: 320KB/WGP) in WMMA-B packed layout, stages h (128KB)
//          in WMMA-A packed layout each step via GLOBAL_LOAD_ASYNC_TO_LDS_B128
//          (ASYNCcnt), runs 256 f32 WMMAs per wave, applies tanh/leaky, and
//          ping-pongs h through d_ws with one device-wide barrier per step
//          (agent-scope release/acquire fences -> global_wb / global_inv).

typedef __attribute__((ext_vector_type(2))) float v2f;
typedef __attribute__((ext_vector_type(8))) float v8f;

#define B_   32
#define T_   1000
#define DIN  64
#define H_   1024
#define NWG  32          // 32 workgroups * 32 columns = H
#define HBUF_FLOATS (B_ * H_)   // 32768 floats = 128KB per h buffer

__device__ __forceinline__ float my_tanhf(float x) {
#if __has_builtin(__builtin_amdgcn_tanhf)
  return __builtin_amdgcn_tanhf(x);       // v_tanh_f32 (gfx1250)
#elif __has_builtin(__builtin_amdgcn_tanh_f32)
  return __builtin_amdgcn_tanh_f32(x);
#else
  return tanhf(x);
#endif
}

// ---------------------------------------------------------------------------
// Init: zero barrier counter + both h ping-pong buffers (h0 = 0).
// Re-run every call => deterministic across graph replays.
// ---------------------------------------------------------------------------
__global__ void esn_init(float* ws) {
  const int i = blockIdx.x * blockDim.x + threadIdx.x;
  if (i < 64 + 2 * HBUF_FLOATS) ws[i] = 0.0f;
}

// ---------------------------------------------------------------------------
// Phase 1: u_in[bt, h] = sum_d x[bt,d] * W_in[h,d] + b[h]   -> d_out
// One wave computes a 16x64 output strip (4 N-tiles share one A tile).
// ---------------------------------------------------------------------------
__global__ __launch_bounds__(256, 1) void esn_uin(
    const float* __restrict__ x, const float* __restrict__ Win,
    const float* __restrict__ Winb, float* __restrict__ out) {
  const int lane = threadIdx.x & 31;
  const int hi   = lane >> 4;        // 0/1: selects K pair within k-tile
  const int l16  = lane & 15;
  const int wid  = blockIdx.x * 8 + (threadIdx.x >> 5);
  const int mTile = wid >> 4;        // 0..1999 (32000 rows / 16)
  const int ng    = wid & 15;        // 0..15  (64 cols each)
  const int rowBase = mTile * 16;
  const int colBase = ng * 64;

  v8f c[4] = {};
  const float* xrow = x + (long)(rowBase + l16) * DIN + hi * 2;
  const float* wb   = Win + hi * 2;

#pragma unroll
  for (int k0 = 0; k0 < DIN; k0 += 4) {
    const v2f a = *(const v2f*)(xrow + k0);   // A: M=l16, K=k0+2*hi..+1
#pragma unroll
    for (int j = 0; j < 4; ++j) {
      // B[k][n] = Win[n][k], n = colBase + j*16 + l16
      const v2f b = *(const v2f*)(wb + (long)(colBase + j * 16 + l16) * DIN + k0);
      c[j] = __builtin_amdgcn_wmma_f32_16x16x4_f32(false, a, false, b,
                                                   (short)0, c[j], false, false);
    }
  }
#pragma unroll
  for (int j = 0; j < 4; ++j) {
    const int col = colBase + j * 16 + l16;
    const float bias = Winb[col];
#pragma unroll
    for (int v = 0; v < 8; ++v) {
      const long row = rowBase + v + 8 * hi;   // C layout: M = v + 8*hi
      out[row * H_ + col] = c[j][v] + bias;
    }
  }
}

// ---------------------------------------------------------------------------
// Phase 2: persistent scan. grid = 32 WGs x 128 threads (4 waves).
// Wave w: mg = w>>1 (rows mg*16..+15), ngw = w&1 (cols colBase+ngw*16..+15).
// ---------------------------------------------------------------------------
__global__ __launch_bounds__(128, 1) void esn_scan(
    const float* __restrict__ W, float* __restrict__ out,
    float* __restrict__ hbufA, float* __restrict__ hbufB,
    unsigned* __restrict__ counter) {
  // Packed layouts: [group][ktile][lane] -> float2 = exactly one WMMA operand
  // register pair per lane. Conflict-free ds_load_b64 in the inner loop.
  __shared__ v2f wpack[2][256][32];   // 128KB: B operand, this WG's 32 W-rows
  __shared__ v2f hpack[2][256][32];   // 128KB: A operand, full h (32 x 1024)

  const int tid  = threadIdx.x;
  const int lane = tid & 31;
  const int hi   = lane >> 4;
  const int l16  = lane & 15;
  const int wave = tid >> 5;
  const int mg   = wave >> 1;
  const int ngw  = wave & 1;
  const int colBase = blockIdx.x * 32;

  // One-time: load W slice into WMMA-B packed layout.
  // B[k][n] = W[j][k] with j = colBase + g*16 + (lane&15), k = kt*4 + 2*(lane>>4).
  for (int i = tid; i < 2 * 256 * 32; i += 128) {
    const int lp = i & 31;
    const int kt = (i >> 5) & 255;
    const int g  = i >> 13;
    const int n  = colBase + g * 16 + (lp & 15);
    const int k  = kt * 4 + (lp >> 4) * 2;
    wpack[g][kt][lp] = *(const v2f*)(W + (long)n * H_ + k);
  }

  const int jcol  = colBase + ngw * 16 + l16;     // output column (== K index of h)
  const int ktj   = jcol >> 2;                    // packed k-tile of jcol
  const int laneh = 16 * ((l16 >> 1) & 1);        // lane-half from (jcol>>1)&1
  const int slot  = l16 & 1;                      // float2 slot from jcol&1

  // LDS byte offset of hpack (generic LDS pointer: addr[31:0] = LDS offset).
  const unsigned hbase = (unsigned)(uintptr_t)(void*)&hpack[0][0][0];

  for (int t = 0; t < T_; ++t) {
    const float* hrd = (t & 1) ? hbufB : hbufA;
    float*       hwr = (t & 1) ? hbufA : hbufB;

    // Stage h (already stored in packed A layout) into LDS via the CDNA5
    // async memory->LDS path (no VGPR round trip, tracked by ASYNCcnt).
    {
      const float4* src = (const float4*)hrd;
      for (int i = tid; i < HBUF_FLOATS / 4; i += 128) {
        const unsigned lds_addr = hbase + (unsigned)i * 16u;
        const unsigned long long ga = (unsigned long long)(uintptr_t)(src + i);
        asm volatile("global_load_async_to_lds_b128 %0, %1, off"
                     :: "v"(lds_addr), "v"(ga) : "memory");
      }
      asm volatile("s_wait_asynccnt 0x0" ::: "memory");
    }
    __syncthreads();

    // C starts as u_in tile read from d_out (overwritten with h below).
    v8f c;
#pragma unroll
    for (int v = 0; v < 8; ++v)
      c[v] = out[((long)(mg * 16 + v + 8 * hi) * T_ + t) * H_ + jcol];

    // Pull next step's u tile toward this WGP while the WMMA chain runs.
    if (t + 1 < T_) {
#pragma unroll
      for (int v = 0; v < 8; ++v)
        __builtin_prefetch(
            &out[((long)(mg * 16 + v + 8 * hi) * T_ + (t + 1)) * H_ + jcol],
            0, 1);
    }

    const v2f* ap = &hpack[mg][0][lane];
    const v2f* bp = &wpack[ngw][0][lane];
#pragma unroll 8
    for (int kt = 0; kt < 256; ++kt) {   // K = 1024, 4 per WMMA
      c = __builtin_amdgcn_wmma_f32_16x16x4_f32(false, ap[kt * 32], false,
                                                bp[kt * 32], (short)0, c,
                                                false, false);
    }

    // h_new = 0.1*h + 0.9*tanh(u + h@Wt); write to d_out and packed h buffer.
    const float* hp = (const float*)&hpack[mg][ktj][0];
#pragma unroll
    for (int v = 0; v < 8; ++v) {
      const int r = mg * 16 + v + 8 * hi;                 // batch row
      const float hold = hp[((v + 8 * hi) + laneh) * 2 + slot];
      const float hn = 0.1f * hold + 0.9f * my_tanhf(c[v]);
      out[((long)r * T_ + t) * H_ + jcol] = hn;
      hwr[((mg * 256 + ktj) * 32 + (v + 8 * hi) + laneh) * 2 + slot] = hn;
    }

    // Device-wide barrier: release stores, signal, spin, acquire-invalidate.
    __builtin_amdgcn_fence(__ATOMIC_RELEASE, "agent");    // global_wb + waits
    __syncthreads();
    if (tid == 0) {
      __hip_atomic_fetch_add(counter, 1u, __ATOMIC_RELAXED,
                             __HIP_MEMORY_SCOPE_AGENT);
      const unsigned target = (unsigned)NWG * (unsigned)(t + 1);
      while (__hip_atomic_load(counter, __ATOMIC_RELAXED,
                               __HIP_MEMORY_SCOPE_AGENT) < target)
        __builtin_amdgcn_s_sleep(2);
    }
    __syncthreads();
    __builtin_amdgcn_fence(__ATOMIC_ACQUIRE, "agent");    // global_inv
  }
}

// ---------------------------------------------------------------------------
extern "C" void kernel_launch(void* const* d_in, const int* in_sizes, int n_in,
                              void* d_out, int out_size, void* d_ws,
                              size_t ws_size, hipStream_t stream) {
  const float* x    = (const float*)d_in[0];   // [32,1000,64]
  const float* Winw = (const float*)d_in[1];   // [1024,64]
  const float* Winb = (const float*)d_in[2];   // [1024]
  const float* W    = (const float*)d_in[3];   // [1024,1024]
  float* out = (float*)d_out;                  // [32,1000,1024]

  unsigned* counter = (unsigned*)d_ws;
  float* hbufA = (float*)((char*)d_ws + 256);
  float* hbufB = hbufA + HBUF_FLOATS;

  // zero counter + both h buffers (64 + 65536 floats)
  esn_init<<<(64 + 2 * HBUF_FLOATS + 255) / 256, 256, 0, stream>>>((float*)d_ws);
  // u_in GEMM: 2000 M-tiles * 16 N-groups = 32000 waves / 8 per block
  esn_uin<<<4000, 256, 0, stream>>>(x, Winw, Winb, out);
  // persistent scan
  esn_scan<<<NWG, 128, 0, stream>>>(W, out, hbufA, hbufB, counter);
}